// SimCLRInfoNCELoss_74603581931830
// MI455X (gfx1250) — compile-verified
//
#include <hip/hip_runtime.h>
#include <math.h>

typedef __attribute__((ext_vector_type(2))) float v2f;
typedef __attribute__((ext_vector_type(8))) float v8f;
typedef __attribute__((ext_vector_type(4))) int v4i;

constexpr int B = 8192;
constexpr int F = 128;
constexpr int NCHUNK = 16;                          // column chunks for parallelism
constexpr int COLS_PER_CHUNK = B / NCHUNK;          // 512
constexpr int TILES_PER_CHUNK = COLS_PER_CHUNK / 16;// 32
constexpr int ROW_WAVES = B / 16;                   // 512 (one wave owns 16 rows)
constexpr int LDS_PITCH = 132;                      // 128 + 4 pad -> no LDS bank conflicts

#if defined(__gfx1250__) && __has_builtin(__builtin_amdgcn_global_load_async_to_lds_b128)
#define USE_ASYNC_LDS 1
typedef __attribute__((address_space(1))) v4i gv4i;  // global-AS int4
typedef __attribute__((address_space(3))) v4i lv4i;  // LDS-AS int4
#else
#define USE_ASYNC_LDS 0
#endif

__device__ __forceinline__ float wave_sum(float v) {
#pragma unroll
  for (int m = 16; m >= 1; m >>= 1) v += __shfl_xor(v, m, 32);
  return v;
}

// ---------------- Kernel 1: inverse L2 norms of rows of z1 and z2 -----------
__global__ void __launch_bounds__(256)
simclr_norms_kernel(const float* __restrict__ z1, const float* __restrict__ z2,
                    float* __restrict__ inv) {
  const int wid  = blockIdx.x * 8 + (threadIdx.x >> 5);   // 0 .. 2B-1
  const int lane = threadIdx.x & 31;
  const float* src = (wid < B) ? z1 : z2;
  const int row = (wid < B) ? wid : wid - B;
  const float4 q = *(const float4*)(src + row * F + lane * 4);
  float ss = q.x * q.x + q.y * q.y + q.z * q.z + q.w * q.w;
  ss = wave_sum(ss);
  if (lane == 0) inv[wid] = 1.0f / fmaxf(sqrtf(ss), 1e-12f);
}

// --------- Kernel 2: fused fp32-WMMA GEMM + online logsumexp partials -------
// Wave w: rowWave = w % 512 -> rows [16*rowWave, +16); chunk = w / 512 -> cols
// [512*chunk, +512). Block = 8 waves sharing one chunk; z2 16-col tiles staged
// through LDS once per block (8x reuse), via async global->LDS when available.
__global__ void __launch_bounds__(256)
simclr_gemm_lse_kernel(const float* __restrict__ z1, const float* __restrict__ z2,
                       const float* __restrict__ inv,
                       float* __restrict__ pm, float* __restrict__ ps,
                       float* __restrict__ pp) {
  __shared__ float z2t[16 * LDS_PITCH];

  const int tid  = threadIdx.x;
  const int lane = tid & 31;
  const int half = lane >> 4;           // 0: lanes 0-15, 1: lanes 16-31
  const int n    = lane & 15;
  const int w    = blockIdx.x * 8 + (tid >> 5);
  const int chunk   = w / ROW_WAVES;
  const int rowWave = w % ROW_WAVES;
  const int rowBase = rowWave * 16;
  const int colBase = chunk * COLS_PER_CHUNK;

  // Hoist A fragments: 16 rows x 128 K = 32 x v2f per lane (64 VGPRs).
  // A 16x4 fp32 layout: lane holds row M = lane%16, K pair at 2*(lane/16).
  v2f a[32];
  {
    const float* ap = z1 + (rowBase + n) * F + 2 * half;
#pragma unroll
    for (int kk = 0; kk < 32; ++kk) a[kk] = *(const v2f*)(ap + kk * 4);
  }
  float inv1v[8];
#pragma unroll
  for (int v = 0; v < 8; ++v) inv1v[v] = inv[rowBase + v + 8 * half];

  float mrun[8], srun[8], pos[8];
#pragma unroll
  for (int v = 0; v < 8; ++v) { mrun[v] = -__builtin_inff(); srun[v] = 0.f; pos[v] = 0.f; }

  // Per-thread staging coordinates: 8 floats each over the 16x128 tile.
  const int se = tid * 8;
  const int sr = se >> 7, sk = se & 127;

  for (int t = 0; t < TILES_PER_CHUNK; ++t) {
    __syncthreads();
    {
      const float* g = z2 + (colBase + t * 16 + sr) * F + sk;
      float* l = z2t + sr * LDS_PITCH + sk;
#if USE_ASYNC_LDS
      // IOFFSET is added to both the global and LDS addresses -> reuse bases.
      __builtin_amdgcn_global_load_async_to_lds_b128((gv4i*)g, (lv4i*)l, 0, 0);
      __builtin_amdgcn_global_load_async_to_lds_b128((gv4i*)g, (lv4i*)l, 16, 0);
#if __has_builtin(__builtin_amdgcn_s_wait_asynccnt)
      __builtin_amdgcn_s_wait_asynccnt(0);
#endif
#else
      const float4 q0 = ((const float4*)g)[0];
      const float4 q1 = ((const float4*)g)[1];
      ((float4*)l)[0] = q0;
      ((float4*)l)[1] = q1;
#endif
    }
    __syncthreads();

    // Prefetch all B fragments for this tile into VGPRs (one ds clause),
    // then run two independent WMMA accumulation chains back-to-back.
    v2f bfrag[32];
    {
      const float* bp = z2t + n * LDS_PITCH + 2 * half;  // B 4x16: lane = col N
#pragma unroll
      for (int kk = 0; kk < 32; ++kk) bfrag[kk] = *(const v2f*)(bp + kk * 4);
    }
    v8f acc0 = {}, acc1 = {};
#pragma unroll
    for (int kk = 0; kk < 32; kk += 2) {
      acc0 = __builtin_amdgcn_wmma_f32_16x16x4_f32(false, a[kk], false, bfrag[kk],
                                                   (short)0, acc0, false, false);
      acc1 = __builtin_amdgcn_wmma_f32_16x16x4_f32(false, a[kk + 1], false, bfrag[kk + 1],
                                                   (short)0, acc1, false, false);
    }

    // Epilogue: scale by inv norms, online (max, sumexp), catch diagonal.
    const int col = colBase + t * 16 + n;
    const float i2 = inv[B + col];
#pragma unroll
    for (int v = 0; v < 8; ++v) {
      const float sv = (acc0[v] + acc1[v]) * inv1v[v] * i2;  // C: row = v + 8*half
      const int grow = rowBase + v + 8 * half;
      if (col == grow) pos[v] += sv;
      const float nm = fmaxf(mrun[v], sv);
      srun[v] = srun[v] * expf(mrun[v] - nm) + expf(sv - nm);
      mrun[v] = nm;
    }
  }

  // Merge the 16 lanes of each half (each saw cols == n mod 16) per row slot.
#pragma unroll
  for (int bit = 1; bit <= 8; bit <<= 1) {
#pragma unroll
    for (int v = 0; v < 8; ++v) {
      const float mo = __shfl_xor(mrun[v], bit, 32);
      const float so = __shfl_xor(srun[v], bit, 32);
      const float po = __shfl_xor(pos[v], bit, 32);
      const float nm = fmaxf(mrun[v], mo);
      srun[v] = srun[v] * expf(mrun[v] - nm) + so * expf(mo - nm);
      mrun[v] = nm;
      pos[v] += po;
    }
  }
  if (n == 0) {
#pragma unroll
    for (int v = 0; v < 8; ++v) {
      const int idx = chunk * B + rowBase + v + 8 * half;
      pm[idx] = mrun[v];
      ps[idx] = srun[v];
      pp[idx] = pos[v];
    }
  }
}

// ---------------- Kernel 3: merge chunks per row -> rowloss -----------------
__global__ void __launch_bounds__(256)
simclr_row_combine_kernel(const float* __restrict__ pm, const float* __restrict__ ps,
                          const float* __restrict__ pp, float* __restrict__ rowloss) {
  const int row = blockIdx.x * 256 + threadIdx.x;
  float M = -__builtin_inff(), S = 0.f, P = 0.f;
#pragma unroll
  for (int c = 0; c < NCHUNK; ++c) {
    const float mo = pm[c * B + row];
    const float so = ps[c * B + row];
    P += pp[c * B + row];
    const float nm = fmaxf(M, mo);
    S = S * expf(M - nm) + so * expf(mo - nm);
    M = nm;
  }
  rowloss[row] = M + logf(S) - P;
}

// ---------------- Kernel 4: deterministic mean -> scalar --------------------
__global__ void __launch_bounds__(256)
simclr_mean_kernel(const float* __restrict__ rowloss, float* __restrict__ out) {
  __shared__ float red[256];
  float acc = 0.f;
  for (int i = threadIdx.x; i < B; i += 256) acc += rowloss[i];
  red[threadIdx.x] = acc;
  __syncthreads();
  for (int s = 128; s > 0; s >>= 1) {
    if (threadIdx.x < s) red[threadIdx.x] += red[threadIdx.x + s];
    __syncthreads();
  }
  if (threadIdx.x == 0) out[0] = red[0] * (1.0f / (float)B);
}

extern "C" void kernel_launch(void* const* d_in, const int* in_sizes, int n_in,
                              void* d_out, int out_size, void* d_ws, size_t ws_size,
                              hipStream_t stream) {
  const float* z1 = (const float*)d_in[0];
  const float* z2 = (const float*)d_in[1];
  float* w = (float*)d_ws;
  float* inv     = w;                      // 2*B
  float* pm      = inv + 2 * B;            // NCHUNK*B
  float* ps      = pm + NCHUNK * B;        // NCHUNK*B
  float* pp      = ps + NCHUNK * B;        // NCHUNK*B
  float* rowloss = pp + NCHUNK * B;        // B
  float* out = (float*)d_out;

  simclr_norms_kernel<<<(2 * B) / 8, 256, 0, stream>>>(z1, z2, inv);
  simclr_gemm_lse_kernel<<<(ROW_WAVES * NCHUNK) / 8, 256, 0, stream>>>(z1, z2, inv, pm, ps, pp);
  simclr_row_combine_kernel<<<B / 256, 256, 0, stream>>>(pm, ps, pp, rowloss);
  simclr_mean_kernel<<<1, 256, 0, stream>>>(rowloss, out);
}